// IonicPotential_33543694582098
// MI455X (gfx1250) — compile-verified
//
#include <hip/hip_runtime.h>
#include <math.h>

// ---------------------------------------------------------------------------
// CDNA5 (gfx1250) implementation of the ionic-potential reference.
// wave32 everywhere; WMMA f32 16x16x4 used for the k.r phase GEMMs;
// async global->LDS staging (ASYNCcnt) for the per-block k-tables.
// ---------------------------------------------------------------------------

typedef __attribute__((ext_vector_type(2))) float v2f;
typedef __attribute__((ext_vector_type(8))) float v8f;

#define KE        14.3996f
#define ALPHAC    0.3f
#define CUTOFF    6.0f
#define PI_F      3.14159265358979f
#define TWO_OSPI  1.1283791670955126f   // 2/sqrt(pi)
#define KMAX      12
#define NATOM     32768
#define MMOL      8
#define PERM      4096
#define NPAIR     2097152
#define KPAD      448                    // multiple of 16 (464 kvecs pass the cut; first KPAD used)
#define KTILES    (KPAD / 16)

// accumulator slots (floats) inside ws
#define ACC_DIP      0
#define ACC_SELF     1              // M
#define ACC_YREAL    (ACC_SELF + MMOL)
#define ACC_YBORN    (ACC_YREAL + MMOL)
#define ACC_YEW      (ACC_YBORN + MMOL)
#define ACC_SLABFAC  (ACC_YEW + MMOL)
#define ACC_FFM      (ACC_SLABFAC + 1)  // M*3
#define ACC_GRAD     (ACC_FFM + MMOL*3) // M*3
#define ACC_COUNT    (ACC_GRAD + MMOL*3)

// ---------------------------------------------------------------------------
// Async global->LDS copy, wave-wide b128 per lane (tracked by ASYNCcnt).
// LDS operand: generic pointer to __shared__ truncates to LDS byte address
// (ISA 10.2 aperture rules: low 32 address bits kept).
// ---------------------------------------------------------------------------
__device__ __forceinline__ void async_stage_f32(int tid, int nthreads,
                                                float* lds_dst, const float* gsrc,
                                                int nfloats) {
    int nchunks = nfloats >> 2;   // b128 chunks
    for (int c = tid; c < nchunks; c += nthreads) {
        unsigned loff = (unsigned)(uintptr_t)(lds_dst + 4 * c);
        unsigned long long ga = (unsigned long long)(uintptr_t)(gsrc + 4 * c);
        asm volatile("global_load_async_to_lds_b128 %0, %1, off"
                     :: "v"(loff), "v"(ga) : "memory");
    }
}

__device__ __forceinline__ void async_wait_all() {
    asm volatile("s_wait_asynccnt 0" ::: "memory");
}

// ---------------------------------------------------------------------------
__global__ void zero_kernel(float* p, int n) {
    int t = blockIdx.x * blockDim.x + threadIdx.x;
    if (t < n) p[t] = 0.0f;
}

// build kept k-vector integer list (same order as numpy meshgrid ravel)
__global__ void kvec_build_kernel(int* kxky, int* kcnt) {
    if (threadIdx.x == 0 && blockIdx.x == 0) {
        int cnt = 0;
        for (int a = 0; a < 2 * KMAX + 1; ++a) {
            int kx = (a <= KMAX) ? a : -(a - KMAX);
            for (int b = 0; b < 2 * KMAX + 1; ++b) {
                int ky = (b <= KMAX) ? b : -(b - KMAX);
                int nrm = kx * kx + ky * ky;
                if (nrm != 0 && nrm <= KMAX * KMAX + 2) {
                    kxky[2 * cnt + 0] = kx;
                    kxky[2 * cnt + 1] = ky;
                    ++cnt;
                }
            }
        }
        kcnt[0] = cnt;
    }
}

// per-molecule: recip = 2*pi*inv(cell)^T, vbox = |det|
__global__ void recip_setup_kernel(const float* cell, float* recip, float* vbox) {
    int m = threadIdx.x;
    if (m >= MMOL) return;
    const float* c = cell + m * 9;
    float a00 = c[0], a01 = c[1], a02 = c[2];
    float a10 = c[3], a11 = c[4], a12 = c[5];
    float a20 = c[6], a21 = c[7], a22 = c[8];
    float c00 =  (a11 * a22 - a12 * a21);
    float c01 = -(a10 * a22 - a12 * a20);
    float c02 =  (a10 * a21 - a11 * a20);
    float det = a00 * c00 + a01 * c01 + a02 * c02;
    float id  = 1.0f / det;
    float inv[9];
    inv[0] = c00 * id;
    inv[1] = -(a01 * a22 - a02 * a21) * id;
    inv[2] =  (a01 * a12 - a02 * a11) * id;
    inv[3] = c01 * id;
    inv[4] =  (a00 * a22 - a02 * a20) * id;
    inv[5] = -(a00 * a12 - a02 * a10) * id;
    inv[6] = c02 * id;
    inv[7] = -(a00 * a21 - a01 * a20) * id;
    inv[8] =  (a00 * a11 - a01 * a10) * id;
    float* r = recip + m * 9;
    // recip[d][e] = 2*pi * inv[e][d]
    for (int d = 0; d < 3; ++d)
        for (int e = 0; e < 3; ++e)
            r[d * 3 + e] = 2.0f * PI_F * inv[e * 3 + d];
    vbox[m] = fabsf(det);
}

// cartesian kvecs and gaussian/ksq weights (zero-padded beyond kcnt)
__global__ void kv_kernel(const int* kxky, const int* kcnt, const float* recip,
                          float* kv, float* kw) {
    int t = blockIdx.x * blockDim.x + threadIdx.x;
    if (t >= MMOL * KPAD) return;
    int m = t / KPAD, k = t % KPAD;
    float kvx = 0.f, kvy = 0.f, kvz = 0.f, w = 0.f;
    if (k < kcnt[0]) {
        float fx = (float)kxky[2 * k + 0];
        float fy = (float)kxky[2 * k + 1];
        const float* r = recip + m * 9;
        kvx = fx * r[0] + fy * r[3];
        kvy = fx * r[1] + fy * r[4];
        kvz = fx * r[2] + fy * r[5];
        float ksq = kvx * kvx + kvy * kvy + kvz * kvz;
        if (ksq > 0.f) w = __expf(-0.25f * ksq / ALPHAC) / ksq;
    }
    kv[t * 3 + 0] = kvx;
    kv[t * 3 + 1] = kvy;
    kv[t * 3 + 2] = kvz;
    kw[t] = w;
}

// R_shift, dipole sum(q*z), per-molecule self term sum(q^2)
__global__ void build_rshift_kernel(const float* q, const float* R, const float* shift,
                                    const int* idx_m, const int* is_film,
                                    float* rshift, float* acc) {
    __shared__ float s_dip;
    __shared__ float s_self[MMOL];
    int tid = threadIdx.x;
    if (tid == 0) s_dip = 0.f;
    if (tid < MMOL) s_self[tid] = 0.f;
    __syncthreads();
    int n = blockIdx.x * blockDim.x + tid;
    int m = idx_m[n];
    float f = (float)is_film[n];
    float x = R[n * 3 + 0] + shift[m * 3 + 0] * f;
    float y = R[n * 3 + 1] + shift[m * 3 + 1] * f;
    float z = R[n * 3 + 2] + shift[m * 3 + 2] * f;
    rshift[n * 3 + 0] = x;
    rshift[n * 3 + 1] = y;
    rshift[n * 3 + 2] = z;
    float qn = q[n];
    atomicAdd(&s_dip, qn * z);
    atomicAdd(&s_self[m], qn * qn);
    __syncthreads();
    if (tid == 0) atomicAdd(&acc[ACC_DIP], s_dip);
    if (tid < MMOL) atomicAdd(&acc[ACC_SELF + tid], s_self[tid]);
}

// real-space pair energies + per-atom gradient of (real + born)
__global__ void pair_kernel(const float* q, const float* rshift, const float* offs,
                            const float* r0_ij, const float* n_ij,
                            const int* idx_i, const int* idx_j, const int* idx_m,
                            float* G, float* acc) {
    __shared__ float b_real[MMOL];
    __shared__ float b_born[MMOL];
    int tid = threadIdx.x;
    if (tid < MMOL) { b_real[tid] = 0.f; b_born[tid] = 0.f; }
    __syncthreads();

    int p = blockIdx.x * blockDim.x + tid;
    __builtin_prefetch(&offs[3 * (p + 4096)], 0, 0);  // global_prefetch_b8
    int i = idx_i[p], j = idx_j[p];
    int m = idx_m[i];
    float ux = rshift[j * 3 + 0] - rshift[i * 3 + 0] + offs[p * 3 + 0];
    float uy = rshift[j * 3 + 1] - rshift[i * 3 + 1] + offs[p * 3 + 1];
    float uz = rshift[j * 3 + 2] - rshift[i * 3 + 2] + offs[p * 3 + 2];
    float d = sqrtf(ux * ux + uy * uy + uz * uz);
    if (d < CUTOFF && d > 0.f) {
        float a   = sqrtf(ALPHAC);
        float ad  = a * d;
        float e   = __expf(-ad * ad);
        float er  = erfcf(ad);
        float invd = 1.0f / d;
        float frcut = erfcf(a * CUTOFF) / CUTOFF;
        float qq = q[i] * q[j];
        float pot = qq * (er * invd - frcut);
        float nn  = n_ij[p];
        float r0  = r0_ij[p];
        float B   = fabsf(qq) * __powf(r0, nn - 1.0f) / nn;
        float born = B * (__powf(d, -nn) - __powf(CUTOFF, -nn));
        atomicAdd(&b_real[m], 0.5f * KE * pot);
        atomicAdd(&b_born[m], 0.5f * KE * born);
        // d/dd of pair energy (0.5*KE prefactor included)
        float f1 = -qq * (a * TWO_OSPI * e * invd + er * invd * invd);
        float g1 = -nn * B * __powf(d, -nn - 1.0f);
        float dphi = 0.5f * KE * (f1 + g1);
        float gx = dphi * ux * invd;
        float gy = dphi * uy * invd;
        float gz = dphi * uz * invd;
        atomicAdd(&G[i * 3 + 0], -gx);
        atomicAdd(&G[i * 3 + 1], -gy);
        atomicAdd(&G[i * 3 + 2], -gz);
        atomicAdd(&G[j * 3 + 0],  gx);
        atomicAdd(&G[j * 3 + 1],  gy);
        atomicAdd(&G[j * 3 + 2],  gz);
    }
    __syncthreads();
    if (tid < MMOL) {
        atomicAdd(&acc[ACC_YREAL + tid], b_real[tid]);
        atomicAdd(&acc[ACC_YBORN + tid], b_born[tid]);
    }
}

// ---------------------------------------------------------------------------
// Structure factors via WMMA: kdp[16 atoms, 16 kvecs] = A(16x4) x B(4x16)
// blocks: M*KTILES, 256 threads (8 waves); each wave strides atom-tiles.
// ---------------------------------------------------------------------------
__global__ void sf_kernel(const float* rshift, const float* q, const float* kv,
                          float* sre, float* sim) {
    int m    = blockIdx.x / KTILES;
    int kt   = blockIdx.x % KTILES;
    int wave = threadIdx.x >> 5;
    int lane = threadIdx.x & 31;
    int col  = lane & 15;
    int half = lane >> 4;
    int kbase = kt * 16;

    // B matrix: B[dim, kvec]; VGPR0 = dims {0,2}, VGPR1 = dims {1,3}
    const float* kvm = kv + (m * KPAD + kbase + col) * 3;
    v2f B;
    B.x = half ? kvm[2] : kvm[0];
    B.y = half ? 0.0f   : kvm[1];

    float acc_re = 0.f, acc_im = 0.f;
    for (int tile = wave; tile < PERM / 16; tile += 8) {
        int abase = m * PERM + tile * 16;
        int row = lane & 15;
        const float* rp = rshift + (abase + row) * 3;
        v2f A;
        A.x = half ? rp[2] : rp[0];
        A.y = half ? 0.0f  : rp[1];
        v8f C = {};
        C = __builtin_amdgcn_wmma_f32_16x16x4_f32(
                false, A, false, B, (short)0, C, false, false);
        // D layout: element (atom = abase + v + 8*half, kvec = kbase + col)
        float pr = 0.f, pi = 0.f;
#pragma unroll
        for (int v = 0; v < 8; ++v) {
            float qa = q[abase + v + 8 * half];
            float ph = C[v];
            float sn, cs;
            __sincosf(ph, &sn, &cs);
            pr += qa * cs;
            pi += qa * sn;
        }
        acc_re += pr;
        acc_im += pi;
    }
    // fold the two halves (same k column)
    acc_re += __shfl_xor(acc_re, 16, 32);
    acc_im += __shfl_xor(acc_im, 16, 32);
    if (lane < 16) {
        atomicAdd(&sre[m * KPAD + kbase + lane], acc_re);
        atomicAdd(&sim[m * KPAD + kbase + lane], acc_im);
    }
}

// per-molecule reciprocal energy reduction
__global__ void recip_energy_kernel(const float* kw, const float* sre, const float* sim,
                                    const float* vbox, float* acc) {
    __shared__ float sd[256];
    int m = blockIdx.x;
    int tid = threadIdx.x;
    float s = 0.f;
    for (int k = tid; k < KPAD; k += 256) {
        float re = sre[m * KPAD + k];
        float im = sim[m * KPAD + k];
        s += kw[m * KPAD + k] * (re * re + im * im);
    }
    sd[tid] = s;
    __syncthreads();
    for (int off = 128; off > 0; off >>= 1) {
        if (tid < off) sd[tid] += sd[tid + off];
        __syncthreads();
    }
    if (tid == 0) {
        float V = vbox[m];
        acc[ACC_YEW + m] = (2.0f * PI_F / V) * sd[0];
        atomicAdd(&acc[ACC_SLABFAC], KE * 4.0f * PI_F / V);
    }
}

// ---------------------------------------------------------------------------
// Reciprocal per-atom gradient via WMMA phase tiles.
// k-tables (kv, kw, S_re, S_im ~10.75 KB) are staged once per block into LDS
// with async b128 loads (ASYNCcnt), removing 8x-redundant per-wave VMEM reads
// from the hot loop. blocks: M*32; 8 waves, each owns one 16-atom tile.
// ---------------------------------------------------------------------------
__global__ void recip_force_kernel(const float* rshift, const float* q, const float* kv,
                                   const float* kw, const float* sre, const float* sim,
                                   const float* vbox, float* G) {
    __shared__ __align__(16) float s_kv[KPAD * 3];
    __shared__ __align__(16) float s_kw[KPAD];
    __shared__ __align__(16) float s_sre[KPAD];
    __shared__ __align__(16) float s_sim[KPAD];

    int m    = blockIdx.x >> 5;
    int tb   = blockIdx.x & 31;
    int tid  = threadIdx.x;
    int wave = tid >> 5;
    int lane = tid & 31;
    int col  = lane & 15;
    int half = lane >> 4;
    int tile = tb * 8 + wave;
    int abase = m * PERM + tile * 16;

    // cooperative async staging of the molecule's k-tables
    async_stage_f32(tid, 256, s_kv,  kv  + m * KPAD * 3, KPAD * 3);
    async_stage_f32(tid, 256, s_kw,  kw  + m * KPAD,     KPAD);
    async_stage_f32(tid, 256, s_sre, sre + m * KPAD,     KPAD);
    async_stage_f32(tid, 256, s_sim, sim + m * KPAD,     KPAD);
    async_wait_all();        // each wave drains its own ASYNCcnt
    __syncthreads();         // publish all waves' staged data

    int row = lane & 15;
    const float* rp = rshift + (abase + row) * 3;
    v2f A;
    A.x = half ? rp[2] : rp[0];
    A.y = half ? 0.0f  : rp[1];

    float ax[8], ay[8], az[8];
#pragma unroll
    for (int v = 0; v < 8; ++v) { ax[v] = 0.f; ay[v] = 0.f; az[v] = 0.f; }

    for (int kt = 0; kt < KTILES; ++kt) {
        int kbase = kt * 16;
        int kk = kbase + col;
        float kx = s_kv[kk * 3 + 0];
        float ky = s_kv[kk * 3 + 1];
        float kz = s_kv[kk * 3 + 2];
        float w  = s_kw[kk];
        float Sre = s_sre[kk], Sim = s_sim[kk];
        v2f B;
        B.x = half ? kz : kx;
        B.y = half ? 0.0f : ky;
        v8f C = {};
        C = __builtin_amdgcn_wmma_f32_16x16x4_f32(
                false, A, false, B, (short)0, C, false, false);
#pragma unroll
        for (int v = 0; v < 8; ++v) {
            float qa = q[abase + v + 8 * half];
            float sn, cs;
            __sincosf(C[v], &sn, &cs);
            float t = w * qa * (Sim * cs - Sre * sn);
            ax[v] += t * kx;
            ay[v] += t * ky;
            az[v] += t * kz;
        }
    }
    float factor = KE * 4.0f * PI_F / vbox[m];
#pragma unroll
    for (int v = 0; v < 8; ++v) {
        float sx = ax[v], sy = ay[v], sz = az[v];
#pragma unroll
        for (int off = 1; off < 16; off <<= 1) {
            sx += __shfl_xor(sx, off, 32);
            sy += __shfl_xor(sy, off, 32);
            sz += __shfl_xor(sz, off, 32);
        }
        if (col == v) {
            int at = abase + v + 8 * half;
            atomicAdd(&G[at * 3 + 0], factor * sx);
            atomicAdd(&G[at * 3 + 1], factor * sy);
            atomicAdd(&G[at * 3 + 2], factor * sz);
        }
    }
}

// film-weighted per-molecule force sum (includes slab z-gradient)
__global__ void film_combine_kernel(const float* q, const int* idx_m, const int* is_film,
                                    const float* G, float* acc) {
    __shared__ float bf[MMOL][3];
    int tid = threadIdx.x;
    if (tid < MMOL) { bf[tid][0] = 0.f; bf[tid][1] = 0.f; bf[tid][2] = 0.f; }
    __syncthreads();
    int n = blockIdx.x * blockDim.x + tid;
    int m = idx_m[n];
    float f = (float)is_film[n];
    float gx = G[n * 3 + 0];
    float gy = G[n * 3 + 1];
    float gz = G[n * 3 + 2] + acc[ACC_SLABFAC] * acc[ACC_DIP] * q[n];
    if (f != 0.f) {
        atomicAdd(&bf[m][0], -gx);
        atomicAdd(&bf[m][1], -gy);
        atomicAdd(&bf[m][2], -gz);
    }
    __syncthreads();
    if (tid < MMOL) {
        atomicAdd(&acc[ACC_FFM + tid * 3 + 0], bf[tid][0]);
        atomicAdd(&acc[ACC_FFM + tid * 3 + 1], bf[tid][1]);
        atomicAdd(&acc[ACC_FFM + tid * 3 + 2], bf[tid][2]);
    }
}

// pair-Hessian vector product: grad_s ||F_m||^2 contributions
__global__ void hvp_pair_kernel(const float* q, const float* rshift, const float* offs,
                                const float* r0_ij, const float* n_ij,
                                const int* idx_i, const int* idx_j, const int* idx_m,
                                const int* is_film, const float* acc, float* acc_mut) {
    __shared__ float bg[MMOL][3];
    int tid = threadIdx.x;
    if (tid < MMOL) { bg[tid][0] = 0.f; bg[tid][1] = 0.f; bg[tid][2] = 0.f; }
    __syncthreads();
    int p = blockIdx.x * blockDim.x + tid;
    int i = idx_i[p], j = idx_j[p];
    float df = (float)(is_film[j] - is_film[i]);
    if (df != 0.f) {
        int m = idx_m[i];
        float ux = rshift[j * 3 + 0] - rshift[i * 3 + 0] + offs[p * 3 + 0];
        float uy = rshift[j * 3 + 1] - rshift[i * 3 + 1] + offs[p * 3 + 1];
        float uz = rshift[j * 3 + 2] - rshift[i * 3 + 2] + offs[p * 3 + 2];
        float d = sqrtf(ux * ux + uy * uy + uz * uz);
        if (d < CUTOFF && d > 0.f) {
            float invd = 1.0f / d;
            float a  = sqrtf(ALPHAC);
            float ad = a * d;
            float e  = __expf(-ad * ad);
            float er = erfcf(ad);
            float qq = q[i] * q[j];
            float nn = n_ij[p];
            float B  = fabsf(qq) * __powf(r0_ij[p], nn - 1.0f) / nn;
            float f1 = -qq * (a * TWO_OSPI * e * invd + er * invd * invd);
            float f2 =  qq * (a * TWO_OSPI * e * (2.f * a * a + 2.f * invd * invd)
                              + 2.f * er * invd * invd * invd);
            float g1 = -nn * B * __powf(d, -nn - 1.0f);
            float g2 = nn * (nn + 1.0f) * B * __powf(d, -nn - 2.0f);
            float phi1 = 0.5f * KE * (f1 + g1);
            float phi2 = 0.5f * KE * (f2 + g2);
            float hx = ux * invd, hy = uy * invd, hz = uz * invd;
            float Fx = acc[ACC_FFM + m * 3 + 0];
            float Fy = acc[ACC_FFM + m * 3 + 1];
            float Fz = acc[ACC_FFM + m * 3 + 2];
            float ud = hx * Fx + hy * Fy + hz * Fz;
            float t  = phi1 * invd;
            float Hx = phi2 * ud * hx + t * (Fx - ud * hx);
            float Hy = phi2 * ud * hy + t * (Fy - ud * hy);
            float Hz = phi2 * ud * hz + t * (Fz - ud * hz);
            float s = -2.0f * df * df;
            atomicAdd(&bg[m][0], s * Hx);
            atomicAdd(&bg[m][1], s * Hy);
            atomicAdd(&bg[m][2], s * Hz);
        }
    }
    __syncthreads();
    if (tid < MMOL) {
        atomicAdd(&acc_mut[ACC_GRAD + tid * 3 + 0], bg[tid][0]);
        atomicAdd(&acc_mut[ACC_GRAD + tid * 3 + 1], bg[tid][1]);
        atomicAdd(&acc_mut[ACC_GRAD + tid * 3 + 2], bg[tid][2]);
    }
}

__global__ void finalize_kernel(const float* acc, const float* vbox, float* out) {
    int t = threadIdx.x;
    if (t < MMOL) {
        float yreal = acc[ACC_YREAL + t];
        float yborn = acc[ACC_YBORN + t];
        float yew   = acc[ACC_YEW + t];
        float self  = sqrtf(ALPHAC / PI_F) * acc[ACC_SELF + t];
        float dip   = acc[ACC_DIP];
        float yslab = KE * (2.0f * PI_F / vbox[t]) * dip * dip;
        float yrecip = KE * (yew - self) + yslab;
        float ycoul  = yreal + yrecip;
        out[t]      = ycoul + yborn;      // y_energy
        out[8 + t]  = ycoul;              // y_coulomb
        out[16 + t] = yborn;              // y_born
        float fx = acc[ACC_FFM + t * 3 + 0];
        float fy = acc[ACC_FFM + t * 3 + 1];
        float fz = acc[ACC_FFM + t * 3 + 2];
        out[24 + t] = fx * fx + fy * fy + fz * fz;  // film_force_norm
    }
    if (t < MMOL * 3) out[32 + t] = acc[ACC_GRAD + t];  // film_norm_grad
}

// ---------------------------------------------------------------------------
extern "C" void kernel_launch(void* const* d_in, const int* in_sizes, int n_in,
                              void* d_out, int out_size, void* d_ws, size_t ws_size,
                              hipStream_t stream) {
    (void)in_sizes; (void)n_in; (void)out_size; (void)ws_size;
    const float* q       = (const float*)d_in[0];
    const float* R       = (const float*)d_in[1];
    const float* shift   = (const float*)d_in[2];
    const float* cell    = (const float*)d_in[3];
    const float* offsets = (const float*)d_in[4];
    const float* r0_ij   = (const float*)d_in[5];
    const float* n_ij    = (const float*)d_in[6];
    const int*   idx_i   = (const int*)d_in[7];
    const int*   idx_j   = (const int*)d_in[8];
    const int*   idx_m   = (const int*)d_in[9];
    const int*   is_film = (const int*)d_in[10];
    float* out = (float*)d_out;

    // workspace carve-up (floats)
    float* ws     = (float*)d_ws;
    float* rshift = ws;                       // N*3
    float* G      = rshift + NATOM * 3;       // N*3
    int*   kxky   = (int*)(G + NATOM * 3);    // KPAD*2
    int*   kcnt   = kxky + KPAD * 2;          // 4
    float* recip  = (float*)(kcnt + 4);       // M*9
    float* vbox   = recip + MMOL * 9;         // M
    float* kv     = vbox + MMOL;              // M*KPAD*3
    float* kw     = kv + MMOL * KPAD * 3;     // M*KPAD
    float* sre    = kw + MMOL * KPAD;         // M*KPAD
    float* sim    = sre + MMOL * KPAD;        // M*KPAD
    float* acc    = sim + MMOL * KPAD;        // ACC_COUNT

    // zero accumulators (G, structure factors, scalar accums)
    zero_kernel<<<(NATOM * 3 + 255) / 256, 256, 0, stream>>>(G, NATOM * 3);
    zero_kernel<<<(MMOL * KPAD * 2 + 255) / 256, 256, 0, stream>>>(sre, MMOL * KPAD * 2);
    zero_kernel<<<1, 128, 0, stream>>>(acc, ACC_COUNT);

    kvec_build_kernel<<<1, 1, 0, stream>>>(kxky, kcnt);
    recip_setup_kernel<<<1, MMOL, 0, stream>>>(cell, recip, vbox);
    kv_kernel<<<(MMOL * KPAD + 255) / 256, 256, 0, stream>>>(kxky, kcnt, recip, kv, kw);
    build_rshift_kernel<<<NATOM / 256, 256, 0, stream>>>(q, R, shift, idx_m, is_film,
                                                         rshift, acc);
    pair_kernel<<<NPAIR / 256, 256, 0, stream>>>(q, rshift, offsets, r0_ij, n_ij,
                                                 idx_i, idx_j, idx_m, G, acc);
    sf_kernel<<<MMOL * KTILES, 256, 0, stream>>>(rshift, q, kv, sre, sim);
    recip_energy_kernel<<<MMOL, 256, 0, stream>>>(kw, sre, sim, vbox, acc);
    recip_force_kernel<<<MMOL * 32, 256, 0, stream>>>(rshift, q, kv, kw, sre, sim,
                                                      vbox, G);
    film_combine_kernel<<<NATOM / 256, 256, 0, stream>>>(q, idx_m, is_film, G, acc);
    hvp_pair_kernel<<<NPAIR / 256, 256, 0, stream>>>(q, rshift, offsets, r0_ij, n_ij,
                                                     idx_i, idx_j, idx_m, is_film,
                                                     acc, acc);
    finalize_kernel<<<1, 64, 0, stream>>>(acc, vbox, out);
}